// SPResGCN_BN_5583457484900
// MI455X (gfx1250) — compile-verified
//
#include <hip/hip_runtime.h>
#include <hip/hip_bf16.h>

typedef __attribute__((ext_vector_type(2))) float v2f;
typedef __attribute__((ext_vector_type(8))) float v8f;

// Problem constants (match reference)
#define K_B    4
#define K_N    8192
#define K_E    131072
#define K_ROWS (K_B * K_N)   // 32768
#define BN_EPSF 1e-5f

// ---------------------------------------------------------------------------
// CDNA5 async global->LDS copy (ASYNCcnt-tracked) via inline asm, with a
// portable fallback for the host pass / non-gfx1250 targets.
// The generic address of a __shared__ object carries the LDS byte offset in
// its low 32 bits (LDS aperture: addr[31:0] == LDS address).
// ---------------------------------------------------------------------------
#if defined(__gfx1250__)
#define USE_ASYNC_LDS 1
#else
#define USE_ASYNC_LDS 0
#endif

__device__ __forceinline__ void async_cp16(const float* g, float* l) {
#if USE_ASYNC_LDS
  const unsigned lds_off = (unsigned)(size_t)l;
  asm volatile("global_load_async_to_lds_b128 %0, %1, off"
               :
               : "v"(lds_off), "v"(g)
               : "memory");
#else
  *(float4*)l = *(const float4*)g;
#endif
}

__device__ __forceinline__ void wait_stage() {
#if USE_ASYNC_LDS
  asm volatile("s_wait_asynccnt 0" ::: "memory");
#endif
  __syncthreads();
}

// ---------------------------------------------------------------------------
// GEMM: out[M x NC] = X[M x K] @ W[K x NC] (+ b0 + b1), fp32 WMMA 16x16x4.
// Block = 256 threads = 8 waves; block tile 128(M) x 16(N); K staged in
// 32-deep double-buffered LDS panels via async global->LDS copies.
//   Xs row stride = 36 floats (144B): 16B-aligned rows for b128 async stores,
//   and 36*r mod 64 = 4*(9r mod 16) covers 16 distinct banks ->
//   conflict-free ds_load_b64 A-fragment reads.
//   Ws (32x16) is shared by all 8 waves (same N tile).
//
// A-frag (16x4 f32, 2 VGPR): lanes 0-15 -> (M=lane, K=k,k+1),
//                            lanes 16-31 -> (M=lane-16, K=k+2,k+3).
// B-frag (4x16 f32, 2 VGPR): lanes 0-15 -> (K=k / k+1, N=lane),
//                            lanes 16-31 -> (K=k+2 / k+3, N=lane-16).
// C/D (16x16 f32, 8 VGPR): VGPR r: lanes 0-15 -> M=r, lanes 16-31 -> M=8+r.
// ---------------------------------------------------------------------------
template <int K, int NC>
__global__ __launch_bounds__(256) void gemm_xw_bias(
    const float* __restrict__ X, const float* __restrict__ W,
    const float* __restrict__ b0, const float* __restrict__ b1,
    float* __restrict__ out) {
  constexpr int KC = 32;          // K panel depth
  constexpr int XSS = KC + 4;     // 36-float row stride (see above)
  constexpr int NCHUNK = K / KC;  // 8

  __shared__ float Xs[2][128 * XSS];  // 2 x 18KB
  __shared__ float Ws[2][KC * 16];    // 2 x 2KB

  const int tid = threadIdx.x;
  const int lane = tid & 31;
  const int wave = tid >> 5;
  const int m0b = blockIdx.x * 128;
  const int n0 = blockIdx.y * 16;
  const int half = lane >> 4;  // 0: lanes 0-15, 1: lanes 16-31
  const int l16 = lane & 15;

  auto stage = [&](int buf, int k0) {
    // X panel: 128 x KC floats = 1024 float4, 4 per thread.
#pragma unroll
    for (int i = 0; i < 4; ++i) {
      const int f = tid + 256 * i;  // float4 id; 8 float4 per row
      const int r = f >> 3;
      const int c4 = f & 7;
      async_cp16(X + (size_t)(m0b + r) * K + k0 + c4 * 4,
                 &Xs[buf][r * XSS + c4 * 4]);
    }
    // W panel: KC x 16 floats = 128 float4 (one per thread, tid < 128).
    if (tid < KC * 4) {
      const int r = tid >> 2;
      const int c4 = tid & 3;
      async_cp16(W + (size_t)(k0 + r) * NC + n0 + c4 * 4,
                 &Ws[buf][r * 16 + c4 * 4]);
    }
  };

  stage(0, 0);
  wait_stage();

  v8f acc = {};
  const int arow = (wave * 16 + l16) * XSS + 2 * half;
  const int boff = (2 * half) * 16 + l16;

  for (int chunk = 0; chunk < NCHUNK; ++chunk) {
    const int cur = chunk & 1;
    if (chunk + 1 < NCHUNK) stage(cur ^ 1, (chunk + 1) * KC);

    const float* xs = &Xs[cur][0];
    const float* ws = &Ws[cur][0];
#pragma unroll
    for (int kk = 0; kk < KC; kk += 4) {
      v2f a = *(const v2f*)(xs + arow + kk);  // ds_load_b64
      v2f b;
      b.x = ws[boff + kk * 16];       // ds_load_b32 (broadcast across waves)
      b.y = ws[boff + kk * 16 + 16];  // ds_load_b32
      acc = __builtin_amdgcn_wmma_f32_16x16x4_f32(false, a, false, b, (short)0,
                                                  acc, false, false);
    }
    wait_stage();  // drain this chunk's async stage, sync buffer swap
  }

  float bias = 0.0f;
  if (b0) bias += b0[n0 + l16];
  if (b1) bias += b1[n0 + l16];

  float* orow = out + (size_t)(m0b + wave * 16 + 8 * half) * NC + n0 + l16;
#pragma unroll
  for (int r = 0; r < 8; ++r) orow[(size_t)r * NC] = acc[r] + bias;
}

// ---------------------------------------------------------------------------
// SpMM scatter: h[b, row[e], :] += ew[e] * support[b, col[e], :]
// One thread per (b, e, 4-channel group); float4 gather + 4 f32 atomics (L2).
// ---------------------------------------------------------------------------
template <int NC>
__global__ __launch_bounds__(256) void spmm_scatter(
    const float* __restrict__ support, const int* __restrict__ row,
    const int* __restrict__ col, const float* __restrict__ ew,
    float* __restrict__ h) {
  constexpr int C4 = NC / 4;
  long long idx = (long long)blockIdx.x * blockDim.x + threadIdx.x;
  if (idx >= (long long)K_B * K_E * C4) return;
  const int c4 = (int)(idx % C4);
  const long long t = idx / C4;
  const int e = (int)(t % K_E);
  const int b = (int)(t / K_E);

  const int r = row[e];
  const int c = col[e];
  const float w = ew[e];

  const float4 s =
      *(const float4*)(support + ((size_t)b * K_N + c) * NC + c4 * 4);
  float* dst = h + ((size_t)b * K_N + r) * NC + c4 * 4;
  atomicAdd(dst + 0, w * s.x);
  atomicAdd(dst + 1, w * s.y);
  atomicAdd(dst + 2, w * s.z);
  atomicAdd(dst + 3, w * s.w);
}

// ---------------------------------------------------------------------------
// BN stats: per-channel sum and sum-of-squares over all B*N rows.
// Block = 256 threads (one channel each), 128 rows per block, coalesced.
// stats[0:NC] = sum, stats[NC:2*NC] = sumsq  (zeroed beforehand).
// ---------------------------------------------------------------------------
template <int NC>
__global__ __launch_bounds__(256) void bn_stats(const float* __restrict__ h,
                                                float* __restrict__ stats) {
  const int c = threadIdx.x;  // blockDim.x == NC == 256
  const int row0 = blockIdx.x * 128;
  float s = 0.0f, sq = 0.0f;
#pragma unroll 4
  for (int r = 0; r < 128; ++r) {
    const float v = h[(size_t)(row0 + r) * NC + c];
    s += v;
    sq += v * v;
  }
  atomicAdd(&stats[c], s);
  atomicAdd(&stats[NC + c], sq);
}

// ---------------------------------------------------------------------------
// BN apply + ReLU (+ optional residual add): in-place over h.
// ---------------------------------------------------------------------------
template <int NC, bool RESID>
__global__ __launch_bounds__(256) void bn_apply_relu(
    float* __restrict__ h, const float* __restrict__ stats,
    const float* __restrict__ gamma, const float* __restrict__ beta,
    const float* __restrict__ resid) {
  const size_t i = (size_t)blockIdx.x * blockDim.x + threadIdx.x;
  if (i >= (size_t)K_ROWS * NC) return;
  const int c = (int)(i % NC);
  const float inv_n = 1.0f / (float)K_ROWS;
  const float mean = stats[c] * inv_n;
  const float var = stats[NC + c] * inv_n - mean * mean;
  float v = (h[i] - mean) * rsqrtf(var + BN_EPSF) * gamma[c] + beta[c];
  v = fmaxf(v, 0.0f);
  if (RESID) v += resid[i];
  h[i] = v;
}

__global__ __launch_bounds__(256) void sigmoid_inplace(float* __restrict__ o,
                                                       size_t n) {
  const size_t i = (size_t)blockIdx.x * blockDim.x + threadIdx.x;
  if (i < n) o[i] = 1.0f / (1.0f + __expf(-o[i]));
}

// ---------------------------------------------------------------------------
extern "C" void kernel_launch(void* const* d_in, const int* in_sizes, int n_in,
                              void* d_out, int out_size, void* d_ws,
                              size_t ws_size, hipStream_t stream) {
  const float* x = (const float*)d_in[0];
  const int* erow = (const int*)d_in[1];
  const int* ecol = (const int*)d_in[2];
  const float* ew = (const float*)d_in[3];
  const float* gc1_w = (const float*)d_in[4];
  const float* gc1_b = (const float*)d_in[5];
  const float* sl1_w = (const float*)d_in[6];
  const float* sl1_b = (const float*)d_in[7];
  const float* bn1_g = (const float*)d_in[8];
  const float* bn1_be = (const float*)d_in[9];
  const float* gc2_w = (const float*)d_in[10];
  const float* gc2_b = (const float*)d_in[11];
  const float* sl2_w = (const float*)d_in[12];
  const float* sl2_b = (const float*)d_in[13];
  const float* bn2_g = (const float*)d_in[14];
  const float* bn2_be = (const float*)d_in[15];
  const float* gc3_w = (const float*)d_in[16];
  const float* gc3_b = (const float*)d_in[17];
  const float* sl3_w = (const float*)d_in[18];
  const float* sl3_b = (const float*)d_in[19];
  float* out = (float*)d_out;

  // Workspace layout (floats): S | H1 | H2 | stats  (~96 MB + 2 KB)
  float* S = (float*)d_ws;                   // support buffer [32768 x 256]
  float* H1 = S + (size_t)K_ROWS * 256;      // layer-1 act / residual y
  float* H2 = H1 + (size_t)K_ROWS * 256;     // layer-2 act
  float* stats = H2 + (size_t)K_ROWS * 256;  // 2*256 floats

  const dim3 blk(256);
  const dim3 gA(K_ROWS / 128, 256 / 16);  // 256 x 16 blocks (NC=256)
  const dim3 gC(K_ROWS / 128, 64 / 16);   // 256 x 4 blocks  (NC=64)
  const int spmm256_blocks = (int)(((long long)K_B * K_E * 64 + 255) / 256);
  const int spmm64_blocks = (int)(((long long)K_B * K_E * 16 + 255) / 256);
  const int ew256_blocks = (int)(((size_t)K_ROWS * 256 + 255) / 256);

  // ---- Layer 1: h1 = spmm(x@W1g) + x@W1s + (b1g + b1s) ----
  gemm_xw_bias<256, 256><<<gA, blk, 0, stream>>>(x, gc1_w, nullptr, nullptr, S);
  gemm_xw_bias<256, 256><<<gA, blk, 0, stream>>>(x, sl1_w, gc1_b, sl1_b, H1);
  spmm_scatter<256><<<spmm256_blocks, blk, 0, stream>>>(S, erow, ecol, ew, H1);
  (void)hipMemsetAsync(stats, 0, 2 * 256 * sizeof(float), stream);
  bn_stats<256><<<K_ROWS / 128, blk, 0, stream>>>(H1, stats);
  bn_apply_relu<256, false>
      <<<ew256_blocks, blk, 0, stream>>>(H1, stats, bn1_g, bn1_be, nullptr);
  // H1 is now y (residual source)

  // ---- Layer 2: h2 = relu(bn(spmm(y@W2g) + y@W2s + b)) + y ----
  gemm_xw_bias<256, 256><<<gA, blk, 0, stream>>>(H1, gc2_w, nullptr, nullptr, S);
  gemm_xw_bias<256, 256><<<gA, blk, 0, stream>>>(H1, sl2_w, gc2_b, sl2_b, H2);
  spmm_scatter<256><<<spmm256_blocks, blk, 0, stream>>>(S, erow, ecol, ew, H2);
  (void)hipMemsetAsync(stats, 0, 2 * 256 * sizeof(float), stream);
  bn_stats<256><<<K_ROWS / 128, blk, 0, stream>>>(H2, stats);
  bn_apply_relu<256, true>
      <<<ew256_blocks, blk, 0, stream>>>(H2, stats, bn2_g, bn2_be, H1);

  // ---- Layer 3: out = sigmoid(spmm(h2@W3g) + h2@W3s + b) ----
  gemm_xw_bias<256, 64><<<gC, blk, 0, stream>>>(H2, gc3_w, nullptr, nullptr, S);
  gemm_xw_bias<256, 64><<<gC, blk, 0, stream>>>(H2, sl3_w, gc3_b, sl3_b, out);
  spmm_scatter<64><<<spmm64_blocks, blk, 0, stream>>>(S, erow, ecol, ew, out);
  const size_t nout = (size_t)K_ROWS * 64;
  sigmoid_inplace<<<(int)((nout + 255) / 256), blk, 0, stream>>>(out, nout);
}